// MultiProxyWassersteinClassifier_48730698940584
// MI455X (gfx1250) — compile-verified
//
#include <hip/hip_runtime.h>
#include <math.h>

typedef float v2f __attribute__((ext_vector_type(2)));
typedef float v8f __attribute__((ext_vector_type(8)));

#define B_   32
#define C_   256
#define N_   784     // 28*28 = 49 tiles of 16
#define KC_  100
#define P_   8
#define TAU_ 0.1f
#define EPS_ 1e-6f
#define ITERS_ 50
#define MU_S (1.0f/784.0f)
#define MU_T (1.0f/8.0f)
#define KSTRIDE 17   // padded row stride for 784x16 K tile in LDS

// ---------------------------------------------------------------------------
// Kernel 1: features (B,C,N) -> L2-normalized fn (B,N,C)  (transpose + norm)
// ---------------------------------------------------------------------------
__global__ void feat_norm_kernel(const float* __restrict__ feat,
                                 float* __restrict__ fn) {
  int wid  = blockIdx.x * (blockDim.x >> 5) + (threadIdx.x >> 5); // 800 waves
  int lane = threadIdx.x & 31;
  int b     = wid / 25;
  int chunk = wid % 25;
  int n     = chunk * 32 + lane;
  if (b >= B_ || n >= N_) return;
  const float* fb = feat + (size_t)b * C_ * N_;
  float ss = 0.f;
  for (int c = 0; c < C_; ++c) {
    float x = fb[c * N_ + n];
    ss += x * x;
  }
  float inv = 1.0f / fmaxf(sqrtf(ss), 1e-8f);
  float* dst = fn + ((size_t)b * N_ + n) * C_;
  for (int c = 0; c < C_; ++c) dst[c] = fb[c * N_ + n] * inv;
}

// ---------------------------------------------------------------------------
// Kernel 2: proxies (Kc,P,C) -> L2-normalized pn (Kc,P,C). One wave per proxy.
// ---------------------------------------------------------------------------
__global__ void proxy_norm_kernel(const float* __restrict__ px,
                                  float* __restrict__ pn) {
  int k = blockIdx.x;          // 100 blocks
  int p = threadIdx.x >> 5;    // 8 waves = 8 proxies
  int lane = threadIdx.x & 31;
  const float* src = px + ((size_t)k * P_ + p) * C_;
  float vals[8];
  float ss = 0.f;
#pragma unroll
  for (int i = 0; i < 8; ++i) {
    vals[i] = src[lane + i * 32];
    ss += vals[i] * vals[i];
  }
#pragma unroll
  for (int off = 16; off > 0; off >>= 1) ss += __shfl_xor(ss, off);
  float inv = 1.0f / fmaxf(sqrtf(ss), 1e-8f);
  float* dst = pn + ((size_t)k * P_ + p) * C_;
#pragma unroll
  for (int i = 0; i < 8; ++i) dst[lane + i * 32] = vals[i] * inv;
}

// ---------------------------------------------------------------------------
// One 16-row sim tile via 64 chained V_WMMA_F32_16X16X4_F32.
// ---------------------------------------------------------------------------
__device__ __forceinline__ v8f wmma_tile(const float* __restrict__ fn,
                                         const float* __restrict__ pnT,
                                         size_t rowbase, int half, int lrow,
                                         int tile) {
  v8f acc = {0.f, 0.f, 0.f, 0.f, 0.f, 0.f, 0.f, 0.f};
  const float* arow = fn + (rowbase + (size_t)tile * 16 + lrow) * C_ + half * 2;
#pragma unroll 8
  for (int s = 0; s < 64; ++s) {
    v2f a = *(const v2f*)(arow + s * 4);
    int kb = s * 4 + half * 2;
    v2f bb;
    bb.x = pnT[(kb + 0) * 16 + lrow];  // B: col n=lrow, k = j + 2*half
    bb.y = pnT[(kb + 1) * 16 + lrow];
    acc = __builtin_amdgcn_wmma_f32_16x16x4_f32(
        /*neg_a=*/false, a, /*neg_b=*/false, bb,
        /*c_mod=*/(short)0, acc, /*reuse_a=*/false, /*reuse_b=*/false);
  }
  return acc;
}

__device__ __forceinline__ void store_K(float* __restrict__ Kls, v8f acc,
                                        int tile, int half, int lrow) {
#pragma unroll
  for (int j = 0; j < 8; ++j) {
    int n = tile * 16 + j + 8 * half;  // C layout: row j + 8*(lane/16)
    Kls[n * KSTRIDE + lrow] = expf((acc[j] - 1.0f) * 10.0f);
  }
}

// ---------------------------------------------------------------------------
// Mega-kernel: one WG per (b, class-pair). WMMA f32 GEMM -> K tile in LDS ->
// 50 Sinkhorn iterations in LDS -> logits.
// Dynamic LDS layout (floats), K tile aliases the proxy tile:
//   Kls  [0, 13328)          = 784 * 17     (pnT [0,4096) lives here pre-GEMM)
//   uls  [13328, 14896)      = 2 * 784
//   vls  [14896, 14912)      = 16
//   part [14912, 15184)      = 16 * 17
//   wred [15184, 15200)      = 16
// Total 15200 floats = 60800 bytes (< 64 KB default dynamic-LDS limit).
// ---------------------------------------------------------------------------
__global__ void __launch_bounds__(256) ot_sinkhorn_kernel(
    const float* __restrict__ fn, const float* __restrict__ pn,
    float* __restrict__ out) {
  extern __shared__ float smem[];
  float* pnT  = smem;                // [c][16] proxy tile (aliased, pre-GEMM)
  float* Kls  = smem;                // [784][17]
  float* uls  = smem + 13328;        // [2][784]
  float* vls  = uls + 2 * N_;        // [16]
  float* part = vls + 16;            // [16][17]
  float* wred = part + 16 * KSTRIDE; // [16]

  const int tid  = threadIdx.x;
  const int lane = tid & 31;
  const int wave = tid >> 5;
  const int b    = blockIdx.x / 50;
  const int kp   = blockIdx.x % 50;
  const int k0   = kp * 2;

  // ---- load normalized proxy tile as pnT[c*16 + col], col = kk*8 + p ----
  for (int idx = tid; idx < 16 * C_; idx += 256) {
    int col = idx & 15;
    int c   = idx >> 4;
    pnT[idx] = pn[(((size_t)(k0 + (col >> 3))) * P_ + (col & 7)) * C_ + c];
  }
  __syncthreads();

  // ---- WMMA f32 16x16x4 GEMM: sim tile (784 x 16), accumulate in regs ----
  // Tiles t = wave + 8*i; i = 0..5 valid for all 8 waves (tile <= 47),
  // i = 6 (tile 48) only for wave 0. Fully unrolled so accs[] stays in VGPRs.
  const int half = lane >> 4;  // 0/1
  const int lrow = lane & 15;
  const size_t rowbase = (size_t)b * N_;
  v8f accs[6];
  v8f acc48;
#pragma unroll
  for (int t = 0; t < 6; ++t)
    accs[t] = wmma_tile(fn, pnT, rowbase, half, lrow, wave + t * 8);
  if (wave == 0) acc48 = wmma_tile(fn, pnT, rowbase, half, lrow, 48);
  __syncthreads();  // all waves done reading pnT; safe to alias with Kls

  // ---- K = exp(-(1 - sim)/tau) into LDS ----
#pragma unroll
  for (int t = 0; t < 6; ++t) store_K(Kls, accs[t], wave + t * 8, half, lrow);
  if (wave == 0) store_K(Kls, acc48, 48, half, lrow);
  if (tid < 16) vls[tid] = 1.0f;  // v0 = ones
  __syncthreads();

  // ---- 50 Sinkhorn iterations, all in LDS ----
  const int col = tid & 15;  // v-step column (kk*8 + p)
  const int sub = tid >> 4;  // v-step row chunk (49 rows each)
  const int grp = col >> 3;  // which u (class) this column uses
  for (int it = 0; it < ITERS_; ++it) {
    float vr[16];
#pragma unroll
    for (int j = 0; j < 16; ++j) vr[j] = vls[j];
    // u = mu_s / (K v + eps), for both classes of this WG
    for (int n = tid; n < N_; n += 256) {
      const float* krow = Kls + n * KSTRIDE;
      float s0 = 0.f, s1 = 0.f;
#pragma unroll
      for (int p = 0; p < 8; ++p) {
        s0 += krow[p]     * vr[p];
        s1 += krow[p + 8] * vr[p + 8];
      }
      uls[n]      = MU_S / (s0 + EPS_);
      uls[N_ + n] = MU_S / (s1 + EPS_);
    }
    __syncthreads();
    // v = mu_t / (K^T u + eps): 16 threads per column, 49 rows each
    {
      float s = 0.f;
      const float* ug = uls + grp * N_;
      int base = sub * 49;
      for (int i = 0; i < 49; ++i) {
        int n = base + i;
        s += Kls[n * KSTRIDE + col] * ug[n];
      }
      part[col * KSTRIDE + sub] = s;
    }
    __syncthreads();
    if (tid < 16) {
      float t = 0.f;
#pragma unroll
      for (int i = 0; i < 16; ++i) t += part[tid * KSTRIDE + i];
      vls[tid] = MU_T / (t + EPS_);
    }
    __syncthreads();
  }

  // ---- logits = sum u*K*v*sim, sim recovered as 1 + tau*ln(K) ----
  float vr[16];
#pragma unroll
  for (int j = 0; j < 16; ++j) vr[j] = vls[j];
  float acc0 = 0.f, acc1 = 0.f;
  for (int n = tid; n < N_; n += 256) {
    const float* krow = Kls + n * KSTRIDE;
    float u0 = uls[n], u1 = uls[N_ + n];
#pragma unroll
    for (int p = 0; p < 8; ++p) {
      float kv0 = krow[p];
      acc0 += u0 * kv0 * vr[p] * (1.0f + TAU_ * logf(kv0));
      float kv1 = krow[p + 8];
      acc1 += u1 * kv1 * vr[p + 8] * (1.0f + TAU_ * logf(kv1));
    }
  }
#pragma unroll
  for (int off = 16; off > 0; off >>= 1) {
    acc0 += __shfl_down(acc0, off);
    acc1 += __shfl_down(acc1, off);
  }
  if (lane == 0) { wred[wave] = acc0; wred[8 + wave] = acc1; }
  __syncthreads();
  if (tid == 0) {
    float s0 = 0.f, s1 = 0.f;
#pragma unroll
    for (int w = 0; w < 8; ++w) { s0 += wred[w]; s1 += wred[8 + w]; }
    out[(size_t)b * KC_ + k0]     = s0;
    out[(size_t)b * KC_ + k0 + 1] = s1;
  }
}

// ---------------------------------------------------------------------------
extern "C" void kernel_launch(void* const* d_in, const int* in_sizes, int n_in,
                              void* d_out, int out_size, void* d_ws,
                              size_t ws_size, hipStream_t stream) {
  (void)in_sizes; (void)n_in; (void)out_size; (void)ws_size;
  const float* feat = (const float*)d_in[0];  // (32,256,28,28) f32
  const float* prox = (const float*)d_in[1];  // (100,8,256)    f32
  float* out = (float*)d_out;                 // (32,100)       f32

  float* fn = (float*)d_ws;               // 32*784*256 floats
  float* pn = fn + (size_t)B_ * N_ * C_;  // 100*8*256 floats

  feat_norm_kernel<<<100, 256, 0, stream>>>(feat, fn);
  proxy_norm_kernel<<<100, 256, 0, stream>>>(prox, pn);

  size_t smem_bytes = 15200 * sizeof(float);  // 60800 B
  ot_sinkhorn_kernel<<<32 * 50, 256, smem_bytes, stream>>>(fn, pn, out);
}